// MIA_31147102830628
// MI455X (gfx1250) — compile-verified
//
#include <hip/hip_runtime.h>
#include <hip/hip_bf16.h>

#define NUM_USERS 100000
#define NUM_ITEMS 50000
#define NODES     (NUM_USERS + NUM_ITEMS)
#define DIM       64
#define N_EDGES   2000000
#define BATCH     8192
#define N_LAYERS  3

typedef __attribute__((ext_vector_type(2))) float v2f;
typedef __attribute__((ext_vector_type(8))) float v8f;

#if __has_builtin(__builtin_amdgcn_wmma_f32_16x16x4_f32)
#define USE_WMMA 1
#endif

// ---- init: cur = acc = concat(user_pref, item_pref) --------------------------
__global__ void init_k(const float4* __restrict__ u4, const float4* __restrict__ i4,
                       float4* __restrict__ cur, float4* __restrict__ acc) {
  int idx = blockIdx.x * blockDim.x + threadIdx.x;
  if (idx >= NODES * (DIM / 4)) return;
  int node = idx >> 4;
  float4 v = (node < NUM_USERS) ? u4[idx] : i4[idx - NUM_USERS * (DIM / 4)];
  cur[idx] = v;
  acc[idx] = v;
}

__global__ void zero_k(float4* __restrict__ p, int n4) {
  int idx = blockIdx.x * blockDim.x + threadIdx.x;
  if (idx < n4) p[idx] = make_float4(0.f, 0.f, 0.f, 0.f);
}

// ---- scatter: next[row] += val * cur[col]; 16 lanes (one float4 each) per edge
__global__ void scatter_k(const int* __restrict__ rows, const int* __restrict__ cols,
                          const float* __restrict__ vals, const float4* __restrict__ cur,
                          float* __restrict__ nxt) {
  int t = blockIdx.x * blockDim.x + threadIdx.x;
  int e = t >> 4;
  if (e >= N_EDGES) return;
  int l = t & 15;
  int r = rows[e];
  int c = cols[e];
  float v = vals[e];
  float4 x = cur[c * (DIM / 4) + l];
  float* dst = nxt + (size_t)r * DIM + l * 4;
  unsafeAtomicAdd(dst + 0, v * x.x);   // global_atomic_add_f32, no return
  unsafeAtomicAdd(dst + 1, v * x.y);
  unsafeAtomicAdd(dst + 2, v * x.z);
  unsafeAtomicAdd(dst + 3, v * x.w);
}

__global__ void accadd_k(float4* __restrict__ acc, const float4* __restrict__ nxt) {
  int idx = blockIdx.x * blockDim.x + threadIdx.x;
  if (idx >= NODES * (DIM / 4)) return;
  float4 a = acc[idx];
  float4 b = nxt[idx];
  a.x += b.x; a.y += b.y; a.z += b.z; a.w += b.w;
  acc[idx] = a;
}

// ---- scoring: one wave32 per 16-row tile; diag((strong-adj) x u^T) via WMMA --
__global__ void score_k(const float* __restrict__ acc, const int* __restrict__ users,
                        const int* __restrict__ adjit, const int* __restrict__ pool,
                        float* __restrict__ partials) {
  int lane = threadIdx.x;            // 0..31
  int tile = blockIdx.x;             // 0..BATCH/16-1
  int row  = tile * 16 + (lane & 15);
  int half = lane >> 4;              // 0: K={k,k+1}, 1: K={k+2,k+3}

  int nu = users[row];
  int na = NUM_USERS + adjit[row];
  int ns = NUM_USERS + pool[row * 2 + 1];   // strong item
  const float2* up = (const float2*)(acc + (size_t)nu * DIM);
  const float2* ap = (const float2*)(acc + (size_t)na * DIM);
  const float2* sp = (const float2*)(acc + (size_t)ns * DIM);

  float diag;
  bool  valid;
#ifdef USE_WMMA
  v8f c = {};
#pragma unroll
  for (int kk = 0; kk < DIM; kk += 4) {
    int p2 = (kk >> 1) + half;       // float2 index of K = kk + 2*half
    float2 s = sp[p2];
    float2 a = ap[p2];
    float2 u = up[p2];
    v2f A; A.x = s.x - a.x; A.y = s.y - a.y;   // A: 16x4 slice of (strong-adj)
    v2f B; B.x = u.x;       B.y = u.y;         // B: 4x16 slice of u^T (col==row)
    c = __builtin_amdgcn_wmma_f32_16x16x4_f32(false, A, false, B,
                                              (short)0, c, false, false);
  }
  // C/D layout: VGPR r -> (M=r, N=lane) for lanes 0-15; (M=r+8, N=lane-16) for 16-31.
  // Diagonal (d,d): d<8 at lane d VGPR d; d>=8 at lane d+16 VGPR d-8.
  int r = lane & 7;
  switch (r) {
    case 0: diag = c[0]; break;
    case 1: diag = c[1]; break;
    case 2: diag = c[2]; break;
    case 3: diag = c[3]; break;
    case 4: diag = c[4]; break;
    case 5: diag = c[5]; break;
    case 6: diag = c[6]; break;
    default: diag = c[7]; break;
  }
  valid = (lane < 8) || (lane >= 24);
#else
  float d = 0.f;
  for (int k2 = 0; k2 < DIM / 2; ++k2) {
    float2 s = sp[k2];
    float2 a = ap[k2];
    float2 u = up[k2];
    d += (s.x - a.x) * u.x + (s.y - a.y) * u.y;
  }
  diag = d;
  valid = (lane < 16);
#endif

  // merged = acc/4 on both sides -> score diff = diag / 16
  float z = diag * (1.0f / ((N_LAYERS + 1) * (N_LAYERS + 1)));
  float spv = fmaxf(z, 0.0f) + log1pf(expf(-fabsf(z)));   // stable softplus
  float val = valid ? spv : 0.0f;
#pragma unroll
  for (int m = 16; m > 0; m >>= 1) val += __shfl_xor(val, m, 32);
  if (lane == 0) partials[tile] = val;
}

__global__ void finalize_k(const float* __restrict__ partials, float* __restrict__ out) {
  __shared__ float s[BATCH / 16];
  int t = threadIdx.x;
  s[t] = partials[t];
  __syncthreads();
  for (int o = (BATCH / 32); o > 0; o >>= 1) {
    if (t < o) s[t] += s[t + o];
    __syncthreads();
  }
  if (t == 0) out[0] = s[0] / (float)BATCH;
}

extern "C" void kernel_launch(void* const* d_in, const int* in_sizes, int n_in,
                              void* d_out, int out_size, void* d_ws, size_t ws_size,
                              hipStream_t stream) {
  const float* user_pref = (const float*)d_in[0];
  const float* item_pref = (const float*)d_in[1];
  const int*   rows      = (const int*)d_in[2];
  const int*   cols      = (const int*)d_in[3];
  const float* vals      = (const float*)d_in[4];
  const int*   users     = (const int*)d_in[5];
  const int*   adjit     = (const int*)d_in[6];
  const int*   pool      = (const int*)d_in[7];
  // d_in[8] (items_weight) unused by the reference loss
  float* out = (float*)d_out;

  const size_t nodeFloats = (size_t)NODES * DIM;
  float* bufA     = (float*)d_ws;
  float* bufB     = bufA + nodeFloats;
  float* acc      = bufB + nodeFloats;
  float* partials = acc + nodeFloats;

  const int tpb = 256;
  const int n4  = NODES * (DIM / 4);
  const int gN  = (n4 + tpb - 1) / tpb;

  init_k<<<gN, tpb, 0, stream>>>((const float4*)user_pref, (const float4*)item_pref,
                                 (float4*)bufA, (float4*)acc);

  float* cur = bufA;
  float* nxt = bufB;
  for (int layer = 0; layer < N_LAYERS; ++layer) {
    zero_k<<<gN, tpb, 0, stream>>>((float4*)nxt, n4);
    long long tthreads = (long long)N_EDGES * 16;
    int gE = (int)((tthreads + tpb - 1) / tpb);
    scatter_k<<<gE, tpb, 0, stream>>>(rows, cols, vals, (const float4*)cur, nxt);
    accadd_k<<<gN, tpb, 0, stream>>>((float4*)acc, (const float4*)nxt);
    float* tmp = cur; cur = nxt; nxt = tmp;
  }

  score_k<<<BATCH / 16, 32, 0, stream>>>(acc, users, adjit, pool, partials);
  finalize_k<<<1, BATCH / 16, 0, stream>>>(partials, out);
}